// SecondaryCapsules_51513837748710
// MI455X (gfx1250) — compile-verified
//
#include <hip/hip_runtime.h>
#include <hip/hip_bf16.h>

#define NUM_ROUTES 1152
#define NUM_CAPS   10
#define IN_CH      8
#define OUT_CH     16
#define BATCH      512
#define BT         32                       // batch tile per workgroup (2 WMMA tiles)
#define NWAVES     16
#define NTHREADS   (NWAVES * 32)            // 512
#define R_PER_WAVE (NUM_ROUTES / NWAVES)    // 72

#define X_RSTRIDE  (BATCH * IN_CH)              // floats per route in x (4096)
#define W_RSTRIDE  (NUM_CAPS * OUT_CH * IN_CH)  // floats per route in W (1280)

typedef __attribute__((ext_vector_type(2))) float v2f;
typedef __attribute__((ext_vector_type(8))) float v8f;

__device__ __forceinline__ v8f wmma4(v2f a, v2f b, v8f c) {
  return __builtin_amdgcn_wmma_f32_16x16x4_f32(false, a, false, b,
                                               (short)0, c, false, false);
}

// Weighted-sum sweep: s_j = sum_r softmax(b)_r * u_hat_r for 32 batches
// (two 16x16 WMMA D-tiles sharing each W load). c_ij folded into A rows,
// so the reduction over r lives in 4 independent chains of 72 WMMAs.
template <bool UNIFORM>
__device__ __forceinline__ void s_sweep(const float* __restrict__ x,
                                        const float* __restrict__ w,
                                        const float* b_lds,
                                        const float* red_mx,
                                        const float* red_inv,
                                        float* s_red,
                                        int c, int b0, int tid) {
  const int wv   = tid >> 5;
  const int lane = tid & 31;
  const int h    = lane >> 4;
  const int mn   = lane & 15;               // A-row = batch (within tile), B-col = o
  const int kb   = 2 * h;
  const int r0   = wv * R_PER_WAVE;

  const float* xp = x + (size_t)(r0 * X_RSTRIDE + (b0 + mn) * IN_CH + kb);
  const float* wp = w + (size_t)(r0 * W_RSTRIDE + (c * OUT_CH + mn) * IN_CH + kb);
  const float* bp = b_lds + r0 * BT + mn;

  const float mx0  = UNIFORM ? 0.f : red_mx[mn];
  const float mx1  = UNIFORM ? 0.f : red_mx[mn + 16];
  const float inv0 = UNIFORM ? (1.0f / (float)NUM_ROUTES) : red_inv[mn];
  const float inv1 = UNIFORM ? (1.0f / (float)NUM_ROUTES) : red_inv[mn + 16];

  v8f a00 = {0.f,0.f,0.f,0.f,0.f,0.f,0.f,0.f};   // tile0 (batches b0..b0+15), k-lo
  v8f a01 = {0.f,0.f,0.f,0.f,0.f,0.f,0.f,0.f};   // tile0, k-hi
  v8f a10 = {0.f,0.f,0.f,0.f,0.f,0.f,0.f,0.f};   // tile1 (batches b0+16..b0+31), k-lo
  v8f a11 = {0.f,0.f,0.f,0.f,0.f,0.f,0.f,0.f};   // tile1, k-hi

#pragma unroll 2
  for (int t = 0; t < R_PER_WAVE; ++t) {
    const float cw0 = UNIFORM ? inv0 : __expf(bp[0]  - mx0) * inv0;
    const float cw1 = UNIFORM ? inv1 : __expf(bp[16] - mx1) * inv1;
    v2f x00 = *(const v2f*)(xp);                 // tile0, k=kb..kb+1
    v2f x01 = *(const v2f*)(xp + 4);             // tile0, k=4+kb..5+kb
    v2f x10 = *(const v2f*)(xp + 16 * IN_CH);    // tile1
    v2f x11 = *(const v2f*)(xp + 16 * IN_CH + 4);
    v2f bb0 = *(const v2f*)(wp);                 // shared W, k-lo
    v2f bb1 = *(const v2f*)(wp + 4);             // shared W, k-hi
    x00 *= cw0; x01 *= cw0;
    x10 *= cw1; x11 *= cw1;
    a00 = wmma4(x00, bb0, a00);
    a01 = wmma4(x01, bb1, a01);
    a10 = wmma4(x10, bb0, a10);
    a11 = wmma4(x11, bb1, a11);
    xp += X_RSTRIDE;
    wp += W_RSTRIDE;
    bp += BT;
  }
#pragma unroll
  for (int j = 0; j < 8; ++j) {
    s_red[(wv * 32 + lane) * 16 + j]     = a00[j] + a01[j];
    s_red[(wv * 32 + lane) * 16 + 8 + j] = a10[j] + a11[j];
  }
}

// Agreement sweep: b[r][batch] += dot_o(u_hat_r[batch,:], v[batch,:]) for both
// batch tiles. v is route-invariant -> hoisted; o lives across lanes in the
// D layout -> 4-step shfl_xor butterfly within 16-lane halves.
__device__ __forceinline__ void a_sweep(const float* __restrict__ x,
                                        const float* __restrict__ w,
                                        const float* v_lds, float* b_lds,
                                        int c, int b0, int tid) {
  const int wv   = tid >> 5;
  const int lane = tid & 31;
  const int h    = lane >> 4;
  const int n    = lane & 15;
  const int kb   = 2 * h;
  const int r0   = wv * R_PER_WAVE;

  const float* xp = x + (size_t)(r0 * X_RSTRIDE + (b0 + n) * IN_CH + kb);
  const float* wp = w + (size_t)(r0 * W_RSTRIDE + (c * OUT_CH + n) * IN_CH + kb);

  float vl0[8], vl1[8];
#pragma unroll
  for (int j = 0; j < 8; ++j) {
    vl0[j] = v_lds[(j + 8 * h) * OUT_CH + n];
    vl1[j] = v_lds[(16 + j + 8 * h) * OUT_CH + n];
  }

#pragma unroll 2
  for (int t = 0; t < R_PER_WAVE; ++t) {
    const int r = r0 + t;
    v2f x00 = *(const v2f*)(xp);
    v2f x01 = *(const v2f*)(xp + 4);
    v2f x10 = *(const v2f*)(xp + 16 * IN_CH);
    v2f x11 = *(const v2f*)(xp + 16 * IN_CH + 4);
    v2f bb0 = *(const v2f*)(wp);
    v2f bb1 = *(const v2f*)(wp + 4);
    v8f a00 = {0.f,0.f,0.f,0.f,0.f,0.f,0.f,0.f};
    v8f a01 = {0.f,0.f,0.f,0.f,0.f,0.f,0.f,0.f};
    v8f a10 = {0.f,0.f,0.f,0.f,0.f,0.f,0.f,0.f};
    v8f a11 = {0.f,0.f,0.f,0.f,0.f,0.f,0.f,0.f};
    a00 = wmma4(x00, bb0, a00);
    a01 = wmma4(x01, bb1, a01);
    a10 = wmma4(x10, bb0, a10);
    a11 = wmma4(x11, bb1, a11);
    float p0[8], p1[8];
#pragma unroll
    for (int j = 0; j < 8; ++j) {
      p0[j] = (a00[j] + a01[j]) * vl0[j];
      p1[j] = (a10[j] + a11[j]) * vl1[j];
    }
#pragma unroll
    for (int j = 0; j < 8; ++j) {
      p0[j] += __shfl_xor(p0[j], 1, 16);
      p0[j] += __shfl_xor(p0[j], 2, 16);
      p0[j] += __shfl_xor(p0[j], 4, 16);
      p0[j] += __shfl_xor(p0[j], 8, 16);
      p1[j] += __shfl_xor(p1[j], 1, 16);
      p1[j] += __shfl_xor(p1[j], 2, 16);
      p1[j] += __shfl_xor(p1[j], 4, 16);
      p1[j] += __shfl_xor(p1[j], 8, 16);
    }
    if (n == 0) {
#pragma unroll
      for (int j = 0; j < 8; ++j) {
        b_lds[r * BT + 8 * h + j]      += p0[j];
        b_lds[r * BT + 16 + 8 * h + j] += p1[j];
      }
    }
    xp += X_RSTRIDE;
    wp += W_RSTRIDE;
  }
}

__global__ __launch_bounds__(NTHREADS)
void caps_route_kernel(const float* __restrict__ x,
                       const float* __restrict__ w,
                       float* __restrict__ out) {
  __shared__ float b_lds[NUM_ROUTES * BT];        // routing logits, 144 KB
  __shared__ float s_red[NWAVES * 32 * 16];       // per-wave WMMA partials, 32 KB
  __shared__ float s_bo[BT * OUT_CH];
  __shared__ float v_lds[BT * OUT_CH];
  __shared__ float red2d[BT * 16];
  __shared__ float red_mx[BT];
  __shared__ float red_inv[BT];

  const int c   = blockIdx.y;
  const int b0  = blockIdx.x * BT;
  const int tid = threadIdx.x;

  {
    float4 z4 = {0.f, 0.f, 0.f, 0.f};
    float4* b4 = (float4*)b_lds;
    for (int i = tid; i < (NUM_ROUTES * BT) / 4; i += NTHREADS) b4[i] = z4;
  }
  __syncthreads();

  // (b, o) ownership for reductions / squash / output: exactly one pair/thread
  const int b        = tid >> 4;          // 0..31
  const int o        = tid & 15;
  const int bb       = b & 15;            // batch within WMMA tile
  const int tile     = b >> 4;            // which 16-batch D-tile
  const int lane_for = o + 16 * (bb >> 3);
  const int jj       = bb & 7;

  for (int it = 0; it < 3; ++it) {
    const bool uniform = (it == 0);
    if (!uniform) {
      // softmax stats over r per batch column: max, then inv(sum(exp))
      const int bcol = tid & 31;           // 32 batch columns
      const int seg  = tid >> 5;           // 16 segments x 72 routes
      float mx = -3.0e38f;
      for (int k = 0; k < NUM_ROUTES / 16; ++k)
        mx = fmaxf(mx, b_lds[(seg * (NUM_ROUTES / 16) + k) * BT + bcol]);
      red2d[bcol * 16 + seg] = mx;
      __syncthreads();
      if (tid < BT) {
        float m2 = -3.0e38f;
        for (int s2 = 0; s2 < 16; ++s2) m2 = fmaxf(m2, red2d[tid * 16 + s2]);
        red_mx[tid] = m2;
      }
      __syncthreads();
      const float mxa = red_mx[bcol];
      float se = 0.f;
      for (int k = 0; k < NUM_ROUTES / 16; ++k)
        se += __expf(b_lds[(seg * (NUM_ROUTES / 16) + k) * BT + bcol] - mxa);
      red2d[bcol * 16 + seg] = se;
      __syncthreads();
      if (tid < BT) {
        float sv = 0.f;
        for (int s2 = 0; s2 < 16; ++s2) sv += red2d[tid * 16 + s2];
        red_inv[tid] = 1.0f / sv;
      }
      __syncthreads();
    }

    if (uniform)
      s_sweep<true>(x, w, b_lds, red_mx, red_inv, s_red, c, b0, tid);
    else
      s_sweep<false>(x, w, b_lds, red_mx, red_inv, s_red, c, b0, tid);
    __syncthreads();

    // reduce wave partials, squash -> v
    float s = 0.f;
#pragma unroll
    for (int wv2 = 0; wv2 < NWAVES; ++wv2)
      s += s_red[(wv2 * 32 + lane_for) * 16 + tile * 8 + jj];
    s_bo[b * OUT_CH + o] = s;
    __syncthreads();
    float n2 = 0.f;
#pragma unroll
    for (int oo = 0; oo < 16; ++oo) {
      const float t2 = s_bo[b * OUT_CH + oo];
      n2 += t2 * t2;
    }
    const float nn = sqrtf(n2);
    const float vv = s * nn / (1.0f + n2);
    if (it == 2) {
      out[(b0 + b) * (NUM_CAPS * OUT_CH) + c * OUT_CH + o] = vv;
    } else {
      v_lds[b * OUT_CH + o] = vv;
    }
    __syncthreads();

    if (it < 2) {
      a_sweep(x, w, v_lds, b_lds, c, b0, tid);
      __syncthreads();
    }
  }
}

extern "C" void kernel_launch(void* const* d_in, const int* in_sizes, int n_in,
                              void* d_out, int out_size, void* d_ws, size_t ws_size,
                              hipStream_t stream) {
  (void)in_sizes; (void)n_in; (void)out_size; (void)d_ws; (void)ws_size;
  const float* x = (const float*)d_in[0];              // (1152, 512, 8)
  const float* w = (const float*)d_in[1];              // (1152, 10, 16, 8)
  float* out = (float*)d_out;                          // (512, 10, 16)
  dim3 grid(BATCH / BT, NUM_CAPS);
  caps_route_kernel<<<grid, NTHREADS, 0, stream>>>(x, w, out);
}